// F_model_15642270892789
// MI455X (gfx1250) — compile-verified
//
#include <hip/hip_runtime.h>
#include <hip/hip_bf16.h>
#include <math.h>

typedef __attribute__((ext_vector_type(2))) float v2f;
typedef __attribute__((ext_vector_type(8))) float v8f;

#define Bsz   16384
#define Dk    1280
#define NOUT  248
#define NPAD  256
#define MBLK  64        // rows per block (4 row-groups of 16)
#define KCHUNK 128
#define NCHUNK (Dk / KCHUNK)
#define XS_STRIDE 132   // 128 + 4 pad -> bank-conflict-free ds_load_b64
#define XS_BUF   (MBLK * XS_STRIDE)

// Async global->LDS path via inline asm (portable across toolchains; the
// clang builtin's prototype differs between them). Device-pass only.
#if defined(__gfx1250__)
#define HAVE_ASYNC_LDS 1
#else
#define HAVE_ASYNC_LDS 0
#endif

// ---------------------------------------------------------------------------
// Pack W_final (D,8), Wp1 (8,D,6), Wp2 (8,6,D,4) into WMMA-B-fragment order:
// flat f = kq*1024 + T*64 + L*2 + e  maps to  B[k][n] with
//   n = T*16 + (L%16),  k = kq*4 + (L/16)*2 + e
// so the GEMM inner loop does one coalesced 8B load per lane per k-step.
// ---------------------------------------------------------------------------
__global__ __launch_bounds__(256) void pack_w_kernel(
    const float* __restrict__ Wf,   // (D, 8)
    const float* __restrict__ Wp1,  // (8, D, 6)
    const float* __restrict__ Wp2,  // (8, 6, D, 4)
    float* __restrict__ Wp)         // (320 * 1024) floats
{
    int f = blockIdx.x * 256 + threadIdx.x;          // < 320*1024 = 327680
    int e  = f & 1;
    int L  = (f >> 1) & 31;
    int T  = (f >> 6) & 15;
    int kq = f >> 10;                                 // 0..319
    int n = T * 16 + (L & 15);
    int k = kq * 4 + ((L >> 4) << 1) + e;             // 0..1279
    float v = 0.0f;
    if (n < 8) {
        v = Wf[k * 8 + n];
    } else if (n < 56) {
        int m = n - 8;                                // i*6 + c
        v = Wp1[(m / 6) * (Dk * 6) + k * 6 + (m % 6)];
    } else if (n < NOUT) {
        int m = n - 56;                               // (i*6+j)*4 + c
        v = Wp2[(m / 4) * (Dk * 4) + k * 4 + (m % 4)];
    }
    Wp[f] = v;
}

// ---------------------------------------------------------------------------
// Fused GEMM (64 rows x 248 cols per block, K=1280 via v_wmma_f32_16x16x4_f32)
// + tree softmax epilogue. Each wave: 2 N-tiles x 4 row-groups; B fragments
// reused 4x. x is staged HBM->LDS with double-buffered async copies
// (global_load_async_to_lds_b128 / ASYNCcnt) on gfx1250.
// ---------------------------------------------------------------------------
__global__ __launch_bounds__(256) void tree_gemm_kernel(
    const float* __restrict__ x,       // (B, D)
    const float* __restrict__ Wp,      // packed weights (320*1024)
    const float* __restrict__ b_final, // (8)
    const float* __restrict__ wu1,     // (8,6)
    const float* __restrict__ b1,      // (8,6)
    const float* __restrict__ wu2,     // (8,6,4)
    const float* __restrict__ b2,      // (8,6,4)
    float* __restrict__ out)           // (B, 248)
{
    // Phase 1: two x-stage buffers (2 * 64 * 132 floats = 67.6 KB).
    // Phase 2: aliased as G staging, 64 x 256 floats (64 KB <= 67.6 KB).
    __shared__ float sm[2 * XS_BUF];

    const int tid   = threadIdx.x;
    const int lane  = tid & 31;
    const int wave  = tid >> 5;            // 0..7
    const int t0    = wave * 2;            // this wave's two N-tiles
    const int t1    = wave * 2 + 1;
    const int row16 = lane & 15;
    const int koff  = (lane >> 4) << 1;    // 0 or 2
    const int mbase = blockIdx.x * MBLK;   // first row of this block's tile

    v8f acc0[4] = {{}, {}, {}, {}};
    v8f acc1[4] = {{}, {}, {}, {}};

    const float* xrowbase = x + (size_t)mbase * Dk;
    const int bo0 = t0 * 64 + lane * 2;
    const int bo1 = t1 * 64 + lane * 2;

    // Issue one K-chunk's x tile as 8 per-thread copies (2048 float4 total).
    auto issue_chunk = [&](int buf, int kc) {
        float* dstbase = &sm[buf * XS_BUF];
        #pragma unroll
        for (int it = 0; it < 8; ++it) {
            int f4  = tid + it * 256;          // 0..2047 float4 slots
            int r   = f4 >> 5;                 // row 0..63
            int seg = f4 & 31;                 // float4 within row
            const float* src = xrowbase + (size_t)r * Dk + kc + seg * 4;
            float*       dst = &dstbase[r * XS_STRIDE + seg * 4];
#if HAVE_ASYNC_LDS
            // GV mode: 64-bit per-lane global address; vDST = LDS byte offset.
            asm volatile("global_load_async_to_lds_b128 %0, %1, off"
                         :: "v"((unsigned)(unsigned long long)dst),
                            "v"((unsigned long long)src)
                         : "memory");
#else
            *(float4*)dst = *(const float4*)src;
#endif
        }
    };

    // One K-chunk of WMMA work on a staged buffer.
    auto compute_chunk = [&](int buf, int kc) {
        const float* xsb = &sm[buf * XS_BUF];
        const float* wpB = Wp + (kc >> 2) * 1024;
        #pragma unroll 4
        for (int kk = 0; kk < KCHUNK; kk += 4) {
            const float* bq = wpB + (kk >> 2) * 1024;
            v2f b0 = *(const v2f*)(bq + bo0);
            v2f b1 = *(const v2f*)(bq + bo1);
            #pragma unroll
            for (int g = 0; g < 4; ++g) {
                v2f a = *(const v2f*)(xsb + (g * 16 + row16) * XS_STRIDE + koff + kk);
                acc0[g] = __builtin_amdgcn_wmma_f32_16x16x4_f32(
                              false, a, false, b0, (short)0, acc0[g], false, false);
                acc1[g] = __builtin_amdgcn_wmma_f32_16x16x4_f32(
                              false, a, false, b1, (short)0, acc1[g], false, false);
            }
        }
    };

#if HAVE_ASYNC_LDS
    // Double-buffered pipeline: async loads complete in order, so after
    // issuing 8 ops for chunk c+1, waiting asynccnt<=8 guarantees chunk c's
    // 8 older ops have landed in LDS.
    issue_chunk(0, 0);
    for (int c = 0; c < NCHUNK; ++c) {
        const int kc = c * KCHUNK;
        if (c < NCHUNK - 1) {
            issue_chunk((c + 1) & 1, kc + KCHUNK);
            asm volatile("s_wait_asynccnt 0x8" ::: "memory");
        } else {
            asm volatile("s_wait_asynccnt 0x0" ::: "memory");
        }
        __syncthreads();                 // all waves' current-buffer data visible
        compute_chunk(c & 1, kc);
        __syncthreads();                 // done reading before buffer is re-filled
    }
#else
    for (int c = 0; c < NCHUNK; ++c) {
        const int kc = c * KCHUNK;
        issue_chunk(0, kc);
        __syncthreads();
        compute_chunk(0, kc);
        __syncthreads();
    }
#endif

    // ---- spill G tiles to LDS (C/D layout: vgpr v -> row v + 8*(lane/16), col lane%16)
    {
        const int r0 = (lane >> 4) << 3;       // 0 or 8
        const int c  = lane & 15;
        #pragma unroll
        for (int g = 0; g < 4; ++g) {
            #pragma unroll
            for (int v = 0; v < 8; ++v) {
                sm[(g * 16 + v + r0) * NPAD + t0 * 16 + c] = acc0[g][v];
                sm[(g * 16 + v + r0) * NPAD + t1 * 16 + c] = acc1[g][v];
            }
        }
    }
    __syncthreads();

    // ---- epilogue: one thread per (row, root-node i); 64*8 = 512 pairs, 2 passes
    #pragma unroll
    for (int it = 0; it < 2; ++it) {
        const int p = tid + it * 256;
        const int r = p >> 3;                  // 0..63
        const int i = p & 7;                   // 0..7
        const float* g = &sm[r * NPAD];
        float* orow = out + (size_t)(mbase + r) * NOUT;

        // root upper (no softmax at root)
        float u1 = g[i] + b_final[i];
        orow[i] = u1;

        // level-1: softmax over 6 children
        float l1[6];
        float mx1 = -3.402823466e38f;
        #pragma unroll
        for (int c = 0; c < 6; ++c) {
            l1[c] = g[8 + i * 6 + c] + u1 * wu1[i * 6 + c] + b1[i * 6 + c];
            mx1 = fmaxf(mx1, l1[c]);
        }
        float s1 = 0.0f;
        #pragma unroll
        for (int c = 0; c < 6; ++c) { l1[c] = __expf(l1[c] - mx1); s1 += l1[c]; }
        const float inv1 = 1.0f / s1;

        #pragma unroll
        for (int j = 0; j < 6; ++j) {
            float u2 = l1[j] * inv1 * u1;
            orow[8 + i * 6 + j] = u2;

            // level-2: softmax over 4 leaves
            const int nb = (i * 6 + j) * 4;
            float l2[4];
            float mx2 = -3.402823466e38f;
            #pragma unroll
            for (int c = 0; c < 4; ++c) {
                l2[c] = g[56 + nb + c] + u2 * wu2[nb + c] + b2[nb + c];
                mx2 = fmaxf(mx2, l2[c]);
            }
            float s2 = 0.0f;
            #pragma unroll
            for (int c = 0; c < 4; ++c) { l2[c] = __expf(l2[c] - mx2); s2 += l2[c]; }
            const float inv2 = u2 / s2;
            #pragma unroll
            for (int c = 0; c < 4; ++c) orow[56 + nb + c] = l2[c] * inv2;
        }
    }
}

extern "C" void kernel_launch(void* const* d_in, const int* in_sizes, int n_in,
                              void* d_out, int out_size, void* d_ws, size_t ws_size,
                              hipStream_t stream) {
    const float* x       = (const float*)d_in[0];
    const float* W_final = (const float*)d_in[1];
    const float* b_final = (const float*)d_in[2];
    const float* Wp1     = (const float*)d_in[3];
    const float* wu1     = (const float*)d_in[4];
    const float* b1      = (const float*)d_in[5];
    const float* Wp2     = (const float*)d_in[6];
    const float* wu2     = (const float*)d_in[7];
    const float* b2      = (const float*)d_in[8];
    float* out = (float*)d_out;

    float* Wp = (float*)d_ws;   // 320*1024 floats = 1.31 MB packed weights

    // Pack weights into WMMA B-fragment order (every call; deterministic).
    pack_w_kernel<<<(320 * 1024) / 256, 256, 0, stream>>>(W_final, Wp1, Wp2, Wp);

    // Fused GEMM + tree epilogue: 16384/64 = 256 row tiles.
    tree_gemm_kernel<<<Bsz / MBLK, 256, 0, stream>>>(
        x, Wp, b_final, wu1, b1, wu2, b2, out);
}